// SoftDTW_62457414418688
// MI455X (gfx1250) — compile-verified
//
#include <hip/hip_runtime.h>

// Soft-DTW (gamma=1) fused kernel for MI455X / gfx1250, wave32.
// B=64, LX=LY=512, DIM=64. One workgroup per batch; 16 waves; 32x32 DP tiles.
// Distance tiles computed with V_WMMA_F32_16X16X4_F32; DP wavefront in
// registers via wave shuffles; tile boundaries triple-buffered in LDS.

#define B_    64
#define LX_   512
#define LY_   512
#define DIM_  64
#define TS    32          // DP tile size (== wave32)
#define NT    16          // 512 / 32 tiles per dimension
#define DSTRIDE 35        // LDS row stride for D tiles: (s-l)+35*l -> 34l+s mod 64 all distinct
#define BIGF  1e30f

typedef float v2f __attribute__((ext_vector_type(2)));
typedef float v8f __attribute__((ext_vector_type(8)));

// LDS layout (floats)
#define SM_X2 0
#define SM_Y2 (SM_X2 + LX_)
#define SM_H  (SM_Y2 + LY_)            // 3 x 512 bottom-row buffers
#define SM_V  (SM_H + 3 * LY_)         // 3 x 512 right-col buffers
#define SM_D  (SM_V + 3 * LX_)         // 16 waves x 32 x DSTRIDE
#define SM_FLOATS (SM_D + NT * TS * DSTRIDE)

__global__ __launch_bounds__(512, 1)
void softdtw_fused(const float* __restrict__ X, const float* __restrict__ Y,
                   float* __restrict__ out) {
  extern __shared__ float sm[];
  float* x2s = sm + SM_X2;
  float* y2s = sm + SM_Y2;
  float* Hb  = sm + SM_H;
  float* Vb  = sm + SM_V;
  float* Dt  = sm + SM_D;

  const int b    = blockIdx.x;
  const int t    = threadIdx.x;
  const int wv   = t >> 5;
  const int lane = t & 31;
  const float* Xb = X + (size_t)b * LX_ * DIM_;
  const float* Yb = Y + (size_t)b * LY_ * DIM_;

  // ---- Phase 0: per-row squared norms into LDS ----
  {
    const float4* xp = (const float4*)(Xb + (size_t)t * DIM_);
    const float4* yp = (const float4*)(Yb + (size_t)t * DIM_);
    float sx = 0.f, sy = 0.f;
#pragma unroll
    for (int i = 0; i < DIM_ / 4; ++i) {
      float4 vx = xp[i], vy = yp[i];
      sx += vx.x * vx.x + vx.y * vx.y + vx.z * vx.z + vx.w * vx.w;
      sy += vy.x * vy.x + vy.y * vy.y + vy.z * vy.z + vy.w * vy.w;
    }
    x2s[t] = sx;
    y2s[t] = sy;
  }
  __syncthreads();

  const int lm = lane & 15;   // M/N within 16x16 WMMA frag
  const int hf = lane >> 4;   // K-half selector for f32 frags

  for (int d = 0; d < 2 * NT - 1; ++d) {
    const int tiLow = (d - (NT - 1) > 0) ? d - (NT - 1) : 0;
    const int tiHigh = (d < NT - 1) ? d : NT - 1;
    const int cnt = tiHigh - tiLow + 1;

    if (wv < cnt) {
      const int ti = tiLow + wv;
      const int tj = d - ti;
      const int gi0 = ti * TS, gj0 = tj * TS;

      // ---- 32x32 distance tile: 2x2 subtiles of V_WMMA_F32_16X16X4_F32 ----
      v8f acc00, acc01, acc10, acc11;
#pragma unroll
      for (int i = 0; i < 8; ++i) {
        acc00[i] = 0.f; acc01[i] = 0.f; acc10[i] = 0.f; acc11[i] = 0.f;
      }
      // A frag (16x4 f32): lane l -> row M=l%16, K = k0 + 2*(l/16) + {0,1}
      // B frag (4x16 f32): lane l -> col N=l%16, K = k0 + 2*(l/16) + {0,1}  (B = Y^T)
      const float* xr0 = Xb + (size_t)(gi0 + lm) * DIM_ + 2 * hf;
      const float* xr1 = Xb + (size_t)(gi0 + 16 + lm) * DIM_ + 2 * hf;
      const float* yr0 = Yb + (size_t)(gj0 + lm) * DIM_ + 2 * hf;
      const float* yr1 = Yb + (size_t)(gj0 + 16 + lm) * DIM_ + 2 * hf;
#pragma unroll
      for (int k0 = 0; k0 < DIM_; k0 += 4) {
        v2f aA = *(const v2f*)(xr0 + k0);
        v2f aB = *(const v2f*)(xr1 + k0);
        v2f bA = *(const v2f*)(yr0 + k0);
        v2f bB = *(const v2f*)(yr1 + k0);
        acc00 = __builtin_amdgcn_wmma_f32_16x16x4_f32(false, aA, false, bA, (short)0, acc00, false, false);
        acc01 = __builtin_amdgcn_wmma_f32_16x16x4_f32(false, aA, false, bB, (short)0, acc01, false, false);
        acc10 = __builtin_amdgcn_wmma_f32_16x16x4_f32(false, aB, false, bA, (short)0, acc10, false, false);
        acc11 = __builtin_amdgcn_wmma_f32_16x16x4_f32(false, aB, false, bB, (short)0, acc11, false, false);
      }

      // Spill D = x2 + y2 - 2*acc to wave-private LDS (conflict-free stride)
      float* Dtw = Dt + wv * (TS * DSTRIDE);
#pragma unroll
      for (int v = 0; v < 8; ++v) {
        const int ilA = v + 8 * hf;  // C/D layout: VGPR v -> M=v (lanes 0-15), M=v+8 (16-31)
        const float xa = x2s[gi0 + ilA];
        const float xc = x2s[gi0 + 16 + ilA];
        const float ya = y2s[gj0 + lm];
        const float yc = y2s[gj0 + 16 + lm];
        Dtw[ilA * DSTRIDE + lm]             = xa + ya - 2.f * acc00[v];
        Dtw[ilA * DSTRIDE + 16 + lm]        = xa + yc - 2.f * acc01[v];
        Dtw[(16 + ilA) * DSTRIDE + lm]      = xc + ya - 2.f * acc10[v];
        Dtw[(16 + ilA) * DSTRIDE + 16 + lm] = xc + yc - 2.f * acc11[v];
      }

      // ---- boundary prefetch (triple-buffered) ----
      const int pPrev  = (d + 2) % 3;  // (d-1) mod 3
      const int pPrev2 = (d + 1) % 3;  // (d-2) mod 3
      const int pCur   = d % 3;
      float Vl = (tj > 0) ? Vb[pPrev * LX_ + gi0 + lane] : BIGF;  // left col R[i, gj0-1]
      float Vd = __shfl_up(Vl, 1);                                // R[i-1, gj0-1]
      float Hreg = (ti > 0) ? Hb[pPrev * LY_ + gj0 + lane] : BIGF; // top row R[gi0-1, j]
      float corner;
      if (ti == 0 && tj == 0)      corner = 0.f;     // virtual start R[-1,-1]
      else if (ti > 0 && tj > 0)   corner = Vb[pPrev2 * LX_ + gi0 - 1];
      else                         corner = BIGF;

      // ---- in-tile wavefront: lane = local row, 63 steps ----
      float r1 = BIGF, r2 = BIGF;
      const bool lastTile = (d == 2 * NT - 2);
#pragma unroll 1
      for (int s = 0; s < 2 * TS - 1; ++s) {
        const int jl = s - lane;
        const bool act = (jl >= 0) && (jl < TS);
        const float up_sh = __shfl_up(r1, 1);
        const float dg_sh = __shfl_up(r2, 1);
        const float topj  = __shfl(Hreg, jl & 31);
        const float topjm = __shfl(Hreg, (jl - 1) & 31);
        const float left = (jl == 0) ? Vl : r1;
        const float dg = (jl == 0) ? ((lane == 0) ? corner : Vd)
                                   : ((lane == 0) ? topjm : dg_sh);
        const float up = (lane == 0) ? topj : up_sh;
        // softmin (gamma = 1): m - log(sum exp(m - x))
        const float mn = fminf(dg, fminf(up, left));
        const float se = __expf(mn - dg) + __expf(mn - up) + __expf(mn - left);
        const int jc = (jl < 0) ? 0 : ((jl > TS - 1) ? TS - 1 : jl);
        const float rc = Dtw[lane * DSTRIDE + jc] + (mn - __logf(se));
        if (act) {
          if (jl == TS - 1) Vb[pCur * LX_ + gi0 + lane] = rc;       // right column
          if (lane == TS - 1) Hb[pCur * LY_ + gj0 + jl] = rc;       // bottom row
          if (lastTile && lane == TS - 1 && jl == TS - 1) out[b] = rc;  // R[511,511]
          r2 = r1;
          r1 = rc;
        }
      }
    }
    __syncthreads();  // advance tile anti-diagonal
  }
}

extern "C" void kernel_launch(void* const* d_in, const int* in_sizes, int n_in,
                              void* d_out, int out_size, void* d_ws, size_t ws_size,
                              hipStream_t stream) {
  (void)in_sizes; (void)n_in; (void)out_size; (void)d_ws; (void)ws_size;
  const float* X = (const float*)d_in[0];
  const float* Y = (const float*)d_in[1];
  float* out = (float*)d_out;

  const int smemBytes = SM_FLOATS * (int)sizeof(float);  // ~88 KB < 320 KB/WGP
  hipFuncSetAttribute(reinterpret_cast<const void*>(softdtw_fused),
                      hipFuncAttributeMaxDynamicSharedMemorySize, smemBytes);
  softdtw_fused<<<dim3(B_), dim3(512), smemBytes, stream>>>(X, Y, out);
}